// CumprodAlpha2TransmittanceModule_10033043603889
// MI455X (gfx1250) — compile-verified
//
#include <hip/hip_runtime.h>
#include <stdint.h>

// ---------------------------------------------------------------------------
// Segmented cumprod of (1 - alpha) over packed rays  (MI455X / gfx1250, wave32)
//
// Memory-bound: ~70-100 MB of traffic @ 23.3 TB/s => a few microseconds.
// alpha (33.5 MB) fits in the 192 MB L2, so the reduce pass warms it and the
// apply pass re-reads it at L2 rate: the 3-pass structure is nearly free and
// is replay-safe (no inter-block spin-waiting).
//
// Flags are NOT materialized globally: ray_start_idx is sorted (256 KB,
// L2-resident), so each tile rebuilds its 4096-bit flag mask in LDS from two
// binary searches + a ~32-ray scatter. Saves 2 kernel launches and ~3 MB.
//
// CDNA5-specific path: the apply pass stages its alpha tile with the gfx1250
// async copy engine (global_load_async_to_lds_b128), builds the flag mask
// while the DMA is in flight, then drains ASYNCcnt (s_wait_asynccnt).
// ---------------------------------------------------------------------------

#define BLOCK 256            // 8 waves of 32
#define WAVES 8
#define VPT   16             // elements per thread (4 x float4)
#define TILE  (BLOCK * VPT)  // 4096 elements / 16 KB per tile
#define FW    (TILE / 32)    // 128 flag words per tile

// segmented combine: (p,f) (+) (q,g) = (g ? q : p*q, f|g)   [non-commutative]

__device__ __forceinline__ void wave_incl_scan(float& p, unsigned& f) {
  const int lane = threadIdx.x & 31;
#pragma unroll
  for (int d = 1; d < 32; d <<= 1) {
    float    op = __shfl_up(p, (unsigned)d, 32);
    unsigned og = (unsigned)__shfl_up((int)f, (unsigned)d, 32);
    if (lane >= d) {
      if (!f) p = op * p;
      f |= og;
    }
  }
}

// Exclusive segmented scan across the whole 256-thread block.
__device__ __forceinline__ void block_seg_exscan(float& p, unsigned& f,
                                                 float* sP, unsigned* sF) {
  const int tid  = threadIdx.x;
  const int lane = tid & 31;
  const int wv   = tid >> 5;
  float ip = p; unsigned iff = f;
  wave_incl_scan(ip, iff);                   // inclusive within wave
  if (lane == 31) { sP[wv] = ip; sF[wv] = iff; }
  __syncthreads();
  if (tid == 0) {                            // serial exclusive scan of 8 waves
    float rp = 1.f; unsigned rf = 0u;
#pragma unroll
    for (int w = 0; w < WAVES; ++w) {
      float tp = sP[w]; unsigned tf = sF[w];
      sP[w] = rp; sF[w] = rf;
      if (tf) rp = tp; else rp = rp * tp;
      rf |= tf;
    }
  }
  __syncthreads();
  float    wp = sP[wv];  unsigned wf = sF[wv];
  float    ep = __shfl_up(ip, 1u, 32);
  unsigned ef = (unsigned)__shfl_up((int)iff, 1u, 32);
  if (lane == 0) { ep = 1.f; ef = 0u; }
  p = ef ? ep : wp * ep;                     // wavePrefix (+) laneExclusive
  f = wf | ef;
  __syncthreads();
}

__device__ __forceinline__ int lower_bound(const int* __restrict__ a, int n, int key) {
  int lo = 0, hi = n;
  while (lo < hi) {
    int mid = (lo + hi) >> 1;
    if (a[mid] < key) lo = mid + 1; else hi = mid;
  }
  return lo;
}

// Rebuild this tile's flag bitmask in LDS from the sorted ray starts.
// bm has FW words; sRange has 2 ints. Ends with __syncthreads().
__device__ __forceinline__ void build_flags(const int* __restrict__ starts, int R,
                                            int tbase, unsigned* bm, int* sRange) {
  const int tid = threadIdx.x;
  if (tid < FW) bm[tid] = 0u;
  if (tid == 0) sRange[0] = lower_bound(starts, R, tbase);
  if (tid == 1) sRange[1] = lower_bound(starts, R, tbase + TILE);
  __syncthreads();
  const int lo = sRange[0], hi = sRange[1];
  for (int j = lo + tid; j < hi; j += BLOCK) {
    int s = starts[j] - tbase;
    atomicOr(&bm[s >> 5], 1u << (s & 31));
  }
  if (tbase == 0 && tid == 0) atomicOr(&bm[0], 1u);  // reference forces flag@0
  __syncthreads();
}

// ------------------------------ kernels ------------------------------------

__global__ void k_tile_reduce(const float* __restrict__ alpha,
                              const int* __restrict__ starts, int R,
                              int n, float* __restrict__ aggP,
                              unsigned* __restrict__ aggF) {
  __shared__ unsigned bm[FW];
  __shared__ float sP[WAVES]; __shared__ unsigned sF[WAVES];
  __shared__ int sRange[2];
  const int tid   = threadIdx.x;
  const int tbase = blockIdx.x * TILE;
  const long long base = (long long)tbase + tid * VPT;
  const bool full = (tbase + TILE) <= n;

  float av[VPT];
  if (full) {
    const float4* g = (const float4*)(alpha + base);
    float4 a0 = g[0], a1 = g[1], a2 = g[2], a3 = g[3];
    av[0]=a0.x;  av[1]=a0.y;  av[2]=a0.z;  av[3]=a0.w;
    av[4]=a1.x;  av[5]=a1.y;  av[6]=a1.z;  av[7]=a1.w;
    av[8]=a2.x;  av[9]=a2.y;  av[10]=a2.z; av[11]=a2.w;
    av[12]=a3.x; av[13]=a3.y; av[14]=a3.z; av[15]=a3.w;
  } else {
#pragma unroll
    for (int k = 0; k < VPT; ++k) {
      long long idx = base + k;
      av[k] = (idx < (long long)n) ? alpha[idx] : 0.f;   // oma=1: identity
    }
  }

  build_flags(starts, R, tbase, bm, sRange);
  const int bi = tid * VPT;
  unsigned bits = (bm[bi >> 5] >> (bi & 31)) & 0xFFFFu;

  float p = 1.f; unsigned f = 0u;
#pragma unroll
  for (int k = 0; k < VPT; ++k) {
    float oma = 1.f - av[k];
    if ((bits >> k) & 1u) { p = oma; f = 1u; } else { p = p * oma; }
  }
  wave_incl_scan(p, f);
  if ((tid & 31) == 31) { sP[tid >> 5] = p; sF[tid >> 5] = f; }
  __syncthreads();
  if (tid == 0) {
    float rp = sP[0]; unsigned rf = sF[0];
#pragma unroll
    for (int w = 1; w < WAVES; ++w) {
      if (sF[w]) rp = sP[w]; else rp = rp * sP[w];
      rf |= sF[w];
    }
    aggP[blockIdx.x] = rp; aggF[blockIdx.x] = rf;
  }
}

// One block scans all tile aggregates; writes carry-in product per tile.
__global__ void k_scan_tiles(const float* __restrict__ aggP,
                             const unsigned* __restrict__ aggF,
                             float* __restrict__ carry, int nt, int chunk) {
  __shared__ float sP[WAVES]; __shared__ unsigned sF[WAVES];
  const int tid = threadIdx.x;
  const int begin = tid * chunk;

  float p = 1.f; unsigned f = 0u;
  for (int k = 0; k < chunk; ++k) {
    int i = begin + k;
    if (i < nt) {
      float q = aggP[i]; unsigned g = aggF[i];
      if (g) p = q; else p = p * q;
      f |= g;
    }
  }
  block_seg_exscan(p, f, sP, sF);
  float run = p;                         // (flags[0]==1 => .f never needed)
  for (int k = 0; k < chunk; ++k) {
    int i = begin + k;
    if (i < nt) {
      carry[i] = run;
      float q = aggP[i]; unsigned g = aggF[i];
      if (g) run = q; else run = run * q;
    }
  }
}

__global__ void k_tile_out(const float* __restrict__ alpha,
                           const int* __restrict__ starts, int R,
                           const float* __restrict__ carry,
                           float* __restrict__ T, int n) {
  __shared__ float sA[TILE];             // 16 KB staging tile
  __shared__ unsigned bm[FW];
  __shared__ float sP[WAVES]; __shared__ unsigned sF[WAVES];
  __shared__ int sRange[2];
  const int tid   = threadIdx.x;
  const int tbase = blockIdx.x * TILE;
  const long long base = (long long)tbase + tid * VPT;
  const bool full = (tbase + TILE) <= n;

  if (full) {
    // 1) kick off the async global->LDS DMA (4 x b128 per lane)
    unsigned lds0 = (unsigned)(unsigned long long)(uintptr_t)sA;  // LDS offset
    const float* gptr = alpha + tbase;
#pragma unroll
    for (int i = 0; i < 4; ++i) {
      unsigned goff = (unsigned)((tid * VPT + i * 4) * 4);  // byte offset
      unsigned loff = lds0 + goff;
      asm volatile("global_load_async_to_lds_b128 %0, %1, %2"
                   :: "v"(loff), "v"(goff), "s"(gptr) : "memory");
    }
  } else {
    for (int k = tid; k < TILE; k += BLOCK) {
      long long idx = (long long)tbase + k;
      sA[k] = (idx < (long long)n) ? alpha[idx] : 0.f;
    }
  }

  // 2) build the flag bitmask while the DMA is in flight
  build_flags(starts, R, tbase, bm, sRange);
  const int bi = tid * VPT;
  unsigned bits = (bm[bi >> 5] >> (bi & 31)) & 0xFFFFu;

  // 3) drain this wave's async copies (each lane reads back its own 16B x4)
  if (full) asm volatile("s_wait_asynccnt 0x0" ::: "memory");
  __syncthreads();

  float av[VPT];
#pragma unroll
  for (int k = 0; k < VPT; ++k) av[k] = sA[tid * VPT + k];

  // thread aggregate
  float p = 1.f; unsigned f = 0u;
#pragma unroll
  for (int k = 0; k < VPT; ++k) {
    float oma = 1.f - av[k];
    if ((bits >> k) & 1u) { p = oma; f = 1u; } else { p = p * oma; }
  }
  block_seg_exscan(p, f, sP, sF);        // exclusive within tile
  float cin = carry[blockIdx.x];
  float run = f ? p : cin * p;           // carry into this thread

  float ov[VPT];
#pragma unroll
  for (int k = 0; k < VPT; ++k) {
    float oma = 1.f - av[k];
    unsigned b = (bits >> k) & 1u;
    ov[k] = b ? 1.f : run;               // exclusive transmittance
    run   = b ? oma : run * oma;         // inclusive product
  }

  if (full) {
    float4* o = (float4*)(T + base);
    o[0] = make_float4(ov[0],  ov[1],  ov[2],  ov[3]);
    o[1] = make_float4(ov[4],  ov[5],  ov[6],  ov[7]);
    o[2] = make_float4(ov[8],  ov[9],  ov[10], ov[11]);
    o[3] = make_float4(ov[12], ov[13], ov[14], ov[15]);
  } else {
#pragma unroll
    for (int k = 0; k < VPT; ++k) {
      long long idx = base + k;
      if (idx < (long long)n) T[idx] = ov[k];
    }
  }
}

// bg_r = inc[end-1] = T[end-1] * (1 - alpha[end-1]); empty segment -> 1.
__global__ void k_bg(const float* __restrict__ alpha,
                     const int* __restrict__ starts, int R, int n,
                     const float* __restrict__ T, float* __restrict__ bg) {
  int r = blockIdx.x * blockDim.x + threadIdx.x;
  if (r >= R) return;
  int s = starts[r];
  int e = (r + 1 < R) ? starts[r + 1] : n;
  float v = 1.f;
  if (e > s) {
    float a = alpha[e - 1];
    v = T[e - 1] * (1.f - a);
  }
  bg[r] = v;
}

// ------------------------------ launcher -----------------------------------

extern "C" void kernel_launch(void* const* d_in, const int* in_sizes, int n_in,
                              void* d_out, int out_size, void* d_ws, size_t ws_size,
                              hipStream_t stream) {
  const float* alpha  = (const float*)d_in[0];
  const int*   starts = (const int*)d_in[1];
  const int n = in_sizes[0];
  const int R = in_sizes[1];

  float* T  = (float*)d_out;       // [n] transmittance
  float* bg = T + n;               // [R] bg transmittance

  const int nt = (n + TILE - 1) / TILE;

  char* ws = (char*)d_ws;
  size_t off = 0;
  auto al = [](size_t x) { return (x + 255) & ~(size_t)255; };
  float*    aggP  = (float*)(ws + off);    off = al(off + (size_t)nt * 4);
  unsigned* aggF  = (unsigned*)(ws + off); off = al(off + (size_t)nt * 4);
  float*    carry = (float*)(ws + off);    off = al(off + (size_t)nt * 4);

  k_tile_reduce<<<nt, BLOCK, 0, stream>>>(alpha, starts, R, n, aggP, aggF);
  int chunk = (nt + BLOCK - 1) / BLOCK;
  k_scan_tiles<<<1, BLOCK, 0, stream>>>(aggP, aggF, carry, nt, chunk);
  k_tile_out<<<nt, BLOCK, 0, stream>>>(alpha, starts, R, carry, T, n);
  k_bg<<<(R + BLOCK - 1) / BLOCK, BLOCK, 0, stream>>>(alpha, starts, R, n, T, bg);
}